// SpatialContextEmbedding_39745627357348
// MI455X (gfx1250) — compile-verified
//
#include <hip/hip_runtime.h>
#include <hip/hip_bf16.h>
#include <math.h>

#define BQ   64
#define NQ   512
#define EDIM 128
#define FEATN 33
#define LARGEF 1000000.0f

typedef float v2f __attribute__((ext_vector_type(2)));
typedef float v8f __attribute__((ext_vector_type(8)));

// ---------------------------------------------------------------------------
// Kernel A: pairwise spatial features.
// 4 lanes per row (each scans 128 of the 512 j's), 64 rows per 256-thread
// block, batch positions + pad mask staged in LDS. Sector classification is
// done with octant compares instead of atan2f (edges are multiples of pi/4),
// removing the transcendental from the O(N^2) hot loop. Partial features are
// combined across the 4 lanes of a row with two wave32 shfl_xor steps.
// grid = B * 8 = 512 blocks.
// ---------------------------------------------------------------------------
__global__ __launch_bounds__(256) void feat_kernel(
    const float* __restrict__ pos,            // (B, N, 2)
    const unsigned char* __restrict__ pad,    // (B, N) bool
    float* __restrict__ feat)                 // (B*N, FEATN)
{
    const int b     = blockIdx.x >> 3;
    const int rbase = (blockIdx.x & 7) * 64;

    __shared__ float sx[NQ];
    __shared__ float sy[NQ];
    __shared__ unsigned char sp[NQ];

    for (int j = threadIdx.x; j < NQ; j += 256) {
        sx[j] = pos[((size_t)b * NQ + j) * 2 + 0];
        sy[j] = pos[((size_t)b * NQ + j) * 2 + 1];
        sp[j] = pad[(size_t)b * NQ + j];
    }
    __syncthreads();

    const int i    = rbase + ((int)threadIdx.x >> 2);
    const int part = (int)threadIdx.x & 3;
    const int j0   = part * 128;

    const float xi = sx[i];
    const float yi = sy[i];

    float cnt[12];
    float scnt[8], smin[8];
#pragma unroll
    for (int t = 0; t < 12; ++t) cnt[t] = 0.0f;
#pragma unroll
    for (int s = 0; s < 8; ++s) { scnt[s] = 0.0f; smin[s] = LARGEF; }
    float dmin_up = LARGEF, dmin_dn = LARGEF, dmin_lat = LARGEF;
    float dsum = 0.0f, vcnt = 0.0f;

    const float TH[4] = {3.0f, 5.0f, 8.0f, 12.0f};

#pragma unroll 2
    for (int jj = 0; jj < 128; ++jj) {
        const int j = j0 + jj;
        // rel = pos[j] - pos[i]
        const float dx = sx[j] - xi;
        const float dy = sy[j] - yi;
        const float dist = sqrtf(dx * dx + dy * dy + 1e-8f);

        const bool valid = (j != i) && (sp[j] == 0);
        const float dm   = valid ? dist : LARGEF;   // dist_masked

        const float ady = fabsf(dy) * 0.5f;
        const bool up  = dx >  ady;
        const bool dn  = dx < -ady;
        const bool lat = (!up) && (!dn);

#pragma unroll
        for (int t = 0; t < 4; ++t) {
            const bool within = (dm < TH[t]);       // valid implied (dm==LARGE otherwise)
            cnt[t * 3 + 0] += (within && up)  ? 1.0f : 0.0f;
            cnt[t * 3 + 1] += (within && dn)  ? 1.0f : 0.0f;
            cnt[t * 3 + 2] += (within && lat) ? 1.0f : 0.0f;
        }

        dmin_up  = fminf(dmin_up,  (valid && up)  ? dm : LARGEF);
        dmin_dn  = fminf(dmin_dn,  (valid && dn)  ? dm : LARGEF);
        dmin_lat = fminf(dmin_lat, (valid && lat) ? dm : LARGEF);

        // Octant classifier == sector of atan2(dy,dx) vs linspace(-pi,pi,9)
        // edges (all edges are multiples of pi/4, half-open [e_s, e_{s+1})).
        const int secPos = (dx > 0.0f) ? ((dy <  dx)  ? 4 : 5)
                                       : ((dy > -dx)  ? 6 : 7);
        const int secNeg = (dx >= 0.0f) ? ((-dy >  dx) ? 2 : 3)
                                        : ((-dy >= -dx) ? 1 : 0);
        const int sec = (dy > 0.0f) ? secPos
                      : (dy < 0.0f) ? secNeg
                      : ((dx >= 0.0f) ? 4 : -1);   // dy==+0: angle 0 -> s4; angle pi -> none

#pragma unroll
        for (int s = 0; s < 8; ++s) {
            const bool ins = valid && (sec == s);
            scnt[s] += ins ? 1.0f : 0.0f;
            smin[s]  = ins ? fminf(smin[s], dm) : smin[s];
        }

        dsum += valid ? dist : 0.0f;
        vcnt += valid ? 1.0f : 0.0f;
    }

    // Combine the 4 partial scans of this row (lanes part^1, part^2 within wave32).
#define COMB_ADD(v) { v += __shfl_xor(v, 1, 32); v += __shfl_xor(v, 2, 32); }
#define COMB_MIN(v) { v = fminf(v, __shfl_xor(v, 1, 32)); v = fminf(v, __shfl_xor(v, 2, 32)); }
#pragma unroll
    for (int t = 0; t < 12; ++t) COMB_ADD(cnt[t]);
    COMB_MIN(dmin_up); COMB_MIN(dmin_dn); COMB_MIN(dmin_lat);
#pragma unroll
    for (int s = 0; s < 8; ++s) { COMB_ADD(scnt[s]); COMB_MIN(smin[s]); }
    COMB_ADD(dsum); COMB_ADD(vcnt);
#undef COMB_ADD
#undef COMB_MIN

    if (part == 0) {
        float* fo = feat + (size_t)(b * NQ + i) * FEATN;
#pragma unroll
        for (int t = 0; t < 12; ++t) fo[t] = cnt[t];
        fo[12] = fminf(dmin_up  * 0.1f, 1.0f);
        fo[13] = fminf(dmin_dn  * 0.1f, 1.0f);
        fo[14] = fminf(dmin_lat * 0.1f, 1.0f);
#pragma unroll
        for (int s = 0; s < 8; ++s) {
            fo[15 + 2 * s] = scnt[s];
            fo[16 + 2 * s] = fminf(smin[s] * 0.1f, 1.0f);
        }
        // n_neighbors = count within max thresh (classes partition all pairs)
        fo[31] = cnt[9] + cnt[10] + cnt[11];
        const float mean_d = dsum / fmaxf(vcnt, 1.0f) * 0.1f;
        fo[32] = fminf(mean_d, 1.0f);
    }
}

// ---------------------------------------------------------------------------
// Kernel B: h = feat @ W1 + b1 ; LayerNorm ; exact GELU.
// One wave32 per row, 4 channels per lane, barrier-free shfl_xor butterfly
// reductions for mean / variance. Block = 8 waves = 8 rows; grid = 4096.
// ---------------------------------------------------------------------------
__global__ __launch_bounds__(256) void enc1_kernel(
    const float* __restrict__ feat,   // (B*N, FEATN)
    const float* __restrict__ W1,     // (FEATN, E)
    const float* __restrict__ b1,     // (E)
    const float* __restrict__ gamma,  // (E)
    const float* __restrict__ beta,   // (E)
    float* __restrict__ h)            // (B*N, E)
{
    const int wave = (int)threadIdx.x >> 5;
    const int lane = (int)threadIdx.x & 31;
    const int row  = blockIdx.x * 8 + wave;

    const float* __restrict__ fr = feat + (size_t)row * FEATN;

    float acc[4];
#pragma unroll
    for (int q = 0; q < 4; ++q) acc[q] = b1[lane + 32 * q];

#pragma unroll
    for (int k = 0; k < FEATN; ++k) {
        const float fk = fr[k];           // uniform across wave -> broadcast
#pragma unroll
        for (int q = 0; q < 4; ++q)
            acc[q] += fk * W1[(size_t)k * EDIM + lane + 32 * q];
    }

    // mean over 128 channels
    float s = (acc[0] + acc[1]) + (acc[2] + acc[3]);
#pragma unroll
    for (int off = 16; off > 0; off >>= 1) s += __shfl_xor(s, off, 32);
    const float mu = s * (1.0f / EDIM);

    // variance
    float d[4];
    float vs = 0.0f;
#pragma unroll
    for (int q = 0; q < 4; ++q) { d[q] = acc[q] - mu; vs += d[q] * d[q]; }
#pragma unroll
    for (int off = 16; off > 0; off >>= 1) vs += __shfl_xor(vs, off, 32);
    const float inv_sd = 1.0f / sqrtf(vs * (1.0f / EDIM) + 1e-5f);

#pragma unroll
    for (int q = 0; q < 4; ++q) {
        const int e = lane + 32 * q;
        const float xn = d[q] * inv_sd * gamma[e] + beta[e];
        // exact GELU: 0.5*x*(1+erf(x/sqrt(2)))
        const float g = 0.5f * xn * (1.0f + erff(xn * 0.70710678118f));
        h[(size_t)row * EDIM + e] = g;
    }
}

// ---------------------------------------------------------------------------
// Kernel C: out = h @ W2 + b2 via V_WMMA_F32_16X16X4_F32 (fp32 matrix pipe,
// exact-precision path).  One wave = one 16x16 output tile; K = 128 -> 32
// WMMA steps.  Block = 8 waves = the 8 column tiles of one 16-row stripe.
// Fragment layouts per ISA 7.12.2:
//   A (16x4 f32): lane m = l&15 ; VGPR r holds K = 2*(l>=16)+r
//   B (4x16 f32): lane n = l&15 ; VGPR r holds K = 2*(l>=16)+r
//   C/D (16x16) : lane n = l&15 ; VGPR r holds row m = r + 8*(l>=16)
// ---------------------------------------------------------------------------
__global__ __launch_bounds__(256) void gemm2_kernel(
    const float* __restrict__ h,      // (B*N, E)
    const float* __restrict__ W2,     // (E, E)
    const float* __restrict__ b2,     // (E)
    float* __restrict__ out)          // (B*N, E)
{
    const int wave = threadIdx.x >> 5;
    const int lane = threadIdx.x & 31;

    const int rowBase = blockIdx.x * 16;   // M tile
    const int colBase = wave * 16;         // N tile

    const int mn    = lane & 15;           // m for A, n for B/C/D
    const int khalf = lane >> 4;           // 0: K pair {0,1}, 1: K pair {2,3}

    v8f c = {};

    const float* __restrict__ Arow = h + (size_t)(rowBase + mn) * EDIM;

#pragma unroll 8
    for (int k = 0; k < EDIM; k += 4) {
        const int ka = k + 2 * khalf;
        v2f a, bm;
        a.x  = Arow[ka];
        a.y  = Arow[ka + 1];
        bm.x = W2[(size_t)ka * EDIM + colBase + mn];
        bm.y = W2[(size_t)(ka + 1) * EDIM + colBase + mn];
        c = __builtin_amdgcn_wmma_f32_16x16x4_f32(
                /*neg_a=*/false, a, /*neg_b=*/false, bm,
                /*c_mod=*/(short)0, c, /*reuse_a=*/false, /*reuse_b=*/false);
    }

    const float bias = b2[colBase + mn];
#pragma unroll
    for (int r = 0; r < 8; ++r) {
        const int m = r + 8 * khalf;
        out[(size_t)(rowBase + m) * EDIM + colBase + mn] = c[r] + bias;
    }
}

// ---------------------------------------------------------------------------
extern "C" void kernel_launch(void* const* d_in, const int* in_sizes, int n_in,
                              void* d_out, int out_size, void* d_ws, size_t ws_size,
                              hipStream_t stream) {
    const float*         pos   = (const float*)d_in[0];          // (B,N,2)
    const unsigned char* pad   = (const unsigned char*)d_in[1];  // (B,N) bool
    const float*         W1    = (const float*)d_in[2];          // (FEAT,E)
    const float*         b1    = (const float*)d_in[3];          // (E)
    const float*         gamma = (const float*)d_in[4];          // (E)
    const float*         beta  = (const float*)d_in[5];          // (E)
    const float*         W2    = (const float*)d_in[6];          // (E,E)
    const float*         b2    = (const float*)d_in[7];          // (E)
    float*               outp  = (float*)d_out;                  // (B,N,E)

    const size_t rows = (size_t)BQ * NQ;                         // 32768
    float* ws_feat = (float*)d_ws;                               // rows * FEATN
    float* ws_h    = ws_feat + rows * FEATN;                     // rows * EDIM

    // A: features. 64 batches x 8 row-tiles, 4 lanes per row.
    feat_kernel<<<BQ * 8, 256, 0, stream>>>(pos, pad, ws_feat);

    // B: GEMM1 + LayerNorm + GELU. One wave32 per row, 8 rows per block.
    enc1_kernel<<<(int)(rows / 8), 256, 0, stream>>>(ws_feat, W1, b1, gamma, beta, ws_h);

    // C: GEMM2 + bias via f32 WMMA. 2048 M-tiles, 8 waves = 8 N-tiles each.
    gemm2_kernel<<<(int)(rows / 16), 256, 0, stream>>>(ws_h, W2, b2, outp);
}